// NonGalerkinLoss1Eig_87041807221106
// MI455X (gfx1250) — compile-verified
//
#include <hip/hip_runtime.h>
#include <math.h>

#define N         256
#define AS        257          // padded LDS row stride (floats): conflict-free column walks
#define NBLK      32           // 8-wide blocks
#define NPAIR     16           // block pairs per tournament round
#define ROUNDS    (NBLK - 1)
#define NSWEEPS   5
#define SUBSWEEPS 3
#define TILE      272          // 16x17 padded subproblem tile
#define KEIG      20
#define CNTR      128
#define MMAXP     12
#define EPSV      0.01f
#define BATCH     512

typedef float v2f __attribute__((ext_vector_type(2)));
typedef float v8f __attribute__((ext_vector_type(8)));

// D(16x16,f32) += A(16x4,f32) * B(4x16,f32)  -- CDNA5 V_WMMA_F32_16X16X4_F32
__device__ __forceinline__ v8f wmma4(v2f a, v2f b, v8f c) {
  return __builtin_amdgcn_wmma_f32_16x16x4_f32(false, a, false, b, (short)0, c,
                                               false, false);
}

__global__ __launch_bounds__(256)
void eig_loss_kernel(const float* __restrict__ inp,
                     const float* __restrict__ outp,
                     float* __restrict__ partial) {
  extern __shared__ float smem[];
  float* A    = smem;                 // 256*257 floats
  float* Usc  = A + N * AS;           // 16 * 272
  float* Ssc  = Usc + NPAIR * TILE;   // 16 * 272
  float* Vs   = Ssc + NPAIR * TILE;   // 20 * 256 (tracked eigvec rows)
  float* phmz = Vs + KEIG * N;        // 256
  float* red  = phmz + N;             // 256
  int*   gidx = (int*)(red + N);      // 256: gathered col index per (pair, n)

  const int tid  = threadIdx.x;
  const int b    = blockIdx.x;
  const int lane = tid & 31;
  const int l16  = lane & 15;
  const int hf   = lane >> 4;          // which half of the wave (WMMA layout)
  const int wv   = tid >> 5;           // wave id, 0..7

  __builtin_prefetch(&inp[b * 4], 0, 0);
  const float phi0 = inp[b * 4 + 0];
  const float phi1 = inp[b * 4 + 1];
  const float phi2 = inp[b * 4 + 2];
  const int   mval = (int)inp[b * 4 + 3];
  const float o0 = outp[b * 3 + 0];
  const float o1 = outp[b * 3 + 1];
  const float o2 = outp[b * 3 + 2];

  // ---------------- Phase A: Pm = Phi^m via per-row band stencil -------------
  // Row i of acc@Phi: new[k] = phi1*old[k] + phi0*old[k+1] + phi2*old[k-1].
  // Row tid has support k in [tid-12, tid+12]; keep a 27-wide band (1 ghost/side).
  float cur[27];
#pragma unroll
  for (int d = 0; d < 27; ++d) cur[d] = 0.f;
  cur[13] = 1.f;                       // identity row
  for (int step = 0; step < MMAXP; ++step) {
    if (step < mval) {                 // uniform per block
      float nxt[27];
      nxt[0] = 0.f; nxt[26] = 0.f;
#pragma unroll
      for (int d = 1; d < 26; ++d) {
        int k = tid - 13 + d;
        float v = phi1 * cur[d] + phi0 * cur[d + 1] + phi2 * cur[d - 1];
        nxt[d] = (k >= 0 && k < N) ? v : 0.f;
      }
#pragma unroll
      for (int d = 1; d < 26; ++d) cur[d] = nxt[d];
    }
  }
  // zero A row + phmz, then scatter band, then symmetrize (tril + tril(-1)^T)
  for (int j = 0; j < AS; ++j) A[tid * AS + j] = 0.f;
  phmz[tid] = 0.f;
  __syncthreads();
#pragma unroll
  for (int d = 1; d < 26; ++d) {
    int k = tid - 13 + d;
    if (k >= 0 && k < N) {
      A[tid * AS + k] = cur[d];
      if (tid == CNTR) phmz[k] = cur[d];   // phmz = Pm[128,:] (pre-symmetrize)
    }
  }
  __syncthreads();
  for (int j = tid + 1; j <= tid + 12 && j < N; ++j)
    A[tid * AS + j] = A[j * AS + tid];     // upper := lower^T
  __syncthreads();
  // V_sub rows = e_{idx_r}, idx = {0, 255, 254, ..., 237}
  for (int o = tid; o < KEIG * N; o += 256) {
    int r = o >> 8, c = o & 255;
    int ir = (r == 0) ? 0 : (N - r);
    Vs[o] = (c == ir) ? 1.f : 0.f;
  }
  __syncthreads();

  // ---------------- Phase B: two-sided block Jacobi (WMMA updates) ----------
  for (int sweep = 0; sweep < NSWEEPS; ++sweep) {
    for (int rnd = 0; rnd < ROUNDS; ++rnd) {
      // chess-tournament pairing of 32 blocks -> gathered index table.
      // Thread tid owns entry (pair t = tid>>4, slot n = tid&15).
      {
        int t = tid >> 4, n = tid & 15;
        int aB, bB;
        if (t == 0) { aB = NBLK - 1; bB = rnd % (NBLK - 1); }
        else {
          aB = (rnd + t) % (NBLK - 1);
          bB = (rnd + (NBLK - 1) - t) % (NBLK - 1);
        }
        int blk = (n < 8) ? aB : bB;         // plain select, no exec dance
        gidx[tid] = blk * 8 + (n & 7);
      }
      __syncthreads();                       // B0

      // --- phase 1: each wave solves two 16x16 subproblems, builds U_t ---
      for (int sp = 0; sp < 2; ++sp) {
        int t = wv * 2 + sp;
        const int* gp = gidx + t * 16;
        float* S = Ssc + t * TILE;
        float* U = Usc + t * TILE;
        for (int e = lane; e < 256; e += 32) {
          int i = e >> 4, j = e & 15;
          S[i * 17 + j] = A[gp[i] * AS + gp[j]];
          U[i * 17 + j] = (i == j) ? 1.f : 0.f;
        }
        for (int ss = 0; ss < SUBSWEEPS; ++ss) {
          for (int i = 0; i < 15; ++i) {
            for (int j = i + 1; j < 16; ++j) {
              float app = S[i * 17 + i], aqq = S[j * 17 + j], apq = S[i * 17 + j];
              if (fabsf(apq) > 1e-12f) {     // uniform across wave (LDS scalar)
                float tau = (aqq - app) / (2.f * apq);
                float tt  = ((tau >= 0.f) ? 1.f : -1.f) /
                            (fabsf(tau) + sqrtf(1.f + tau * tau));
                float c = 1.f / sqrtf(1.f + tt * tt);
                float s = tt * c;
                if (lane < 16) {
                  // rotate rows i,j
                  float si = S[i * 17 + lane], sj = S[j * 17 + lane];
                  S[i * 17 + lane] = c * si - s * sj;
                  S[j * 17 + lane] = s * si + c * sj;
                  // then cols i,j (uses updated rows); accumulate U
                  float ci = S[lane * 17 + i], cj = S[lane * 17 + j];
                  S[lane * 17 + i] = c * ci - s * cj;
                  S[lane * 17 + j] = s * ci + c * cj;
                  float ui = U[lane * 17 + i], uj = U[lane * 17 + j];
                  U[lane * 17 + i] = c * ui - s * uj;
                  U[lane * 17 + j] = s * ui + c * uj;
                }
              }
            }
          }
        }
      }
      __syncthreads();                       // B1

      // --- phase 2a: V_sub updates into regs (short live range) ---
      float vreg[KEIG];
#pragma unroll
      for (int it = 0; it < KEIG; ++it) {    // 20*256 == 16 pairs * 20 rows * 16
        int o = tid + it * 256;
        int t = o / 320, w = o - t * 320;
        int r = w >> 4, n = w & 15;
        const int* gp = gidx + t * 16;
        const float* U = Usc + t * TILE;
        float acc = 0.f;
#pragma unroll
        for (int k = 0; k < 16; ++k)
          acc += Vs[r * N + gp[k]] * U[k * 17 + n];
        vreg[it] = acc;
      }
      __syncthreads();                       // B2: all old-Vs reads done

      // --- phase 2b: commit V_sub immediately (frees vreg), column GEMMs ---
#pragma unroll
      for (int it = 0; it < KEIG; ++it) {
        int o = tid + it * 256;
        int t = o / 320, w = o - t * 320;
        int r = w >> 4, n = w & 15;
        Vs[r * N + gidx[t * 16 + n]] = vreg[it];
      }
      // A[:,P_t] <- A[:,P_t] * U_t
      for (int task = wv * 32; task < wv * 32 + 32; ++task) {
        int t = task >> 4, r0 = (task & 15) << 4;
        const int* gp = gidx + t * 16;
        const float* U = Usc + t * TILE;
        const int arow = (r0 + l16) * AS;
        v8f acc = {0.f, 0.f, 0.f, 0.f, 0.f, 0.f, 0.f, 0.f};
#pragma unroll
        for (int kk = 0; kk < 4; ++kk) {
          int kb = kk * 4 + 2 * hf;          // A-op: lanes<16 K={0,1}, else {2,3}
          v2f a, bb;
          a.x  = A[arow + gp[kb + 0]];
          a.y  = A[arow + gp[kb + 1]];
          bb.x = U[(kb + 0) * 17 + l16];     // B-op: row K, col = lane
          bb.y = U[(kb + 1) * 17 + l16];
          acc = wmma4(a, bb, acc);
        }
        const int scol = gp[l16];
#pragma unroll
        for (int r = 0; r < 8; ++r)          // C/D: vgpr r -> row r + 8*half
          A[(r0 + r + 8 * hf) * AS + scol] = acc[r];
      }
      __syncthreads();                       // B3

      // --- phase 3: row GEMMs A[P_t,:] <- U_t^T * A[P_t,:] ---
      for (int task = wv * 32; task < wv * 32 + 32; ++task) {
        int t = task >> 4, c0 = (task & 15) << 4;
        const int* gp = gidx + t * 16;
        const float* U = Usc + t * TILE;
        v8f acc = {0.f, 0.f, 0.f, 0.f, 0.f, 0.f, 0.f, 0.f};
#pragma unroll
        for (int kk = 0; kk < 4; ++kk) {
          int kb = kk * 4 + 2 * hf;
          v2f a, bb;
          a.x  = U[(kb + 0) * 17 + l16];     // (U^T)[m][k] = U[k][m]
          a.y  = U[(kb + 1) * 17 + l16];
          bb.x = A[gp[kb + 0] * AS + c0 + l16];
          bb.y = A[gp[kb + 1] * AS + c0 + l16];
          acc = wmma4(a, bb, acc);
        }
#pragma unroll
        for (int r = 0; r < 8; ++r)
          A[gp[r + 8 * hf] * AS + c0 + l16] = acc[r];
      }
      __syncthreads();                       // B4
    }
  }

  // ---------------- Phase C: rank eigenvalues, compute loss -----------------
  float lam = A[tid * AS + tid];
  red[tid] = lam;
  __syncthreads();
  int rk = 0;
  for (int nu = 0; nu < N; ++nu) {
    float lm = red[nu];
    rk += (lm < lam || (lm == lam && nu < tid)) ? 1 : 0;
  }
  float* ph2  = Ssc;                   // reuse scratch
  float* psel = Usc;
  ph2[tid]  = phmz[rk];                // permute coeffs instead of moving V
  psel[tid] = (rk >= CNTR - 1 && rk <= CNTR + 1)
                  ? ((rk == CNTR - 1) ? o0 : (rk == CNTR) ? o1 : o2)
                  : 0.f;
  __syncthreads();

  float loss = 0.f;
  for (int k = 0; k < KEIG; ++k) {
    float v = Vs[k * N + tid];
    red[tid]  = v * ph2[tid];
    phmz[tid] = v * psel[tid];
    __syncthreads();
    for (int s2 = 128; s2 > 0; s2 >>= 1) {
      if (tid < s2) { red[tid] += red[tid + s2]; phmz[tid] += phmz[tid + s2]; }
      __syncthreads();
    }
    if (tid == 0) {
      float a = red[0], bb = phmz[0];
      float d = a - bb;
      loss += d * d / ((1.f - a) * (1.f - a) + EPSV);
    }
    __syncthreads();
  }
  if (tid == 0) partial[b] = loss;
}

__global__ __launch_bounds__(256)
void reduce_loss_kernel(const float* __restrict__ partial, float* __restrict__ out) {
  __shared__ float s[256];
  int t = threadIdx.x;
  s[t] = partial[t] + partial[t + 256];
  __syncthreads();
  for (int k = 128; k > 0; k >>= 1) {
    if (t < k) s[t] += s[t + k];
    __syncthreads();
  }
  if (t == 0) out[0] = s[0];
}

extern "C" void kernel_launch(void* const* d_in, const int* in_sizes, int n_in,
                              void* d_out, int out_size, void* d_ws, size_t ws_size,
                              hipStream_t stream) {
  (void)in_sizes; (void)n_in; (void)out_size; (void)ws_size;
  const float* inp  = (const float*)d_in[0];   // (512,4): phi0..2, m
  const float* outp = (const float*)d_in[1];   // (512,3)
  float* partial = (float*)d_ws;               // 512 floats
  float* out = (float*)d_out;                  // scalar loss

  const size_t smem =
      (size_t)(N * AS + 2 * NPAIR * TILE + KEIG * N + 2 * N) * sizeof(float) +
      (size_t)N * sizeof(int);                 // 321,536 B (<= 320KB LDS/WGP)
  hipFuncSetAttribute(reinterpret_cast<const void*>(eig_loss_kernel),
                      hipFuncAttributeMaxDynamicSharedMemorySize, (int)smem);
  eig_loss_kernel<<<BATCH, 256, smem, stream>>>(inp, outp, partial);
  reduce_loss_kernel<<<1, 256, 0, stream>>>(partial, out);
}